// SoftDTW_41695542510320
// MI455X (gfx1250) — compile-verified
//
#include <hip/hip_runtime.h>
#include <hip/hip_bf16.h>

// Problem constants (match reference)
#define Bb 32
#define Nn 512
#define Mm 512
#define DF 64
#define BIGF 1.0e10f

// DP staging: 8 diagonals per stage, pad-9 rows (gcd(9,64)=1 -> conflict free)
#define SDEPTH 8
#define SPAD 9

typedef float v2f __attribute__((ext_vector_type(2)));
typedef float v8f __attribute__((ext_vector_type(8)));

// Generic shared-memory pointer -> LDS byte offset.  Per CDNA5 aperture rules
// the LDS flat address is {aperture_hi, offset[31:0]} and hardware adds the
// wave's LDS_BASE to the async op's VDST value, so low 32 bits are exactly the
// group-segment-relative offset.
__device__ __forceinline__ unsigned lds_off(const void* p) {
    return (unsigned)(size_t)p;
}

// GLOBAL_LOAD_ASYNC_TO_LDS_B32 (GV mode): copies a dword from vaddr to LDS offset vdst, per lane,
// tracked by ASYNCcnt.
__device__ __forceinline__ void async_load_f32(unsigned lds_byte, const float* g) {
    asm volatile("global_load_async_to_lds_b32 %0, %1, off"
                 :: "v"(lds_byte), "v"(g) : "memory");
}

__device__ __forceinline__ void wait_asynccnt0() {
    asm volatile("s_wait_asynccnt 0x0" ::: "memory");
}

// ---------------------------------------------------------------------------
// Kernel 1: squared row norms of X and Y.
// ---------------------------------------------------------------------------
__global__ __launch_bounds__(256) void softdtw_norms(
    const float* __restrict__ X, const float* __restrict__ Y,
    float* __restrict__ xn, float* __restrict__ yn)
{
    int idx = blockIdx.x * 256 + threadIdx.x;
    const float* src;
    float* dst;
    if (idx < Bb * Nn) {
        src = X + (size_t)idx * DF;
        dst = xn + idx;
    } else {
        int j = idx - Bb * Nn;
        src = Y + (size_t)j * DF;
        dst = yn + j;
    }
    float s = 0.0f;
#pragma unroll
    for (int k = 0; k < DF; k += 4) {
        float4 v = *(const float4*)(src + k);
        s += v.x * v.x + v.y * v.y + v.z * v.z + v.w * v.w;
    }
    *dst = s;
}

// ---------------------------------------------------------------------------
// Kernel 2: batched GEMM  Dws[b,n,m] = -2 * dot(X[b,n,:], Y[b,m,:])
// via V_WMMA_F32_16X16X4_F32 (fp32 WMMA, 16 MACs over K=64 per 16x16 tile).
// ---------------------------------------------------------------------------
__global__ __launch_bounds__(128) void softdtw_gemm(
    const float* __restrict__ X, const float* __restrict__ Y,
    float* __restrict__ Dws)
{
    const int lane = threadIdx.x & 31;
    const int wave = threadIdx.x >> 5;
    const int b  = blockIdx.z;
    const int bm = blockIdx.y;                 // 16-row tile of X
    const int bn = blockIdx.x * 4 + wave;      // 16-row tile of Y (output cols)
    const int half = lane >> 4;                // 0: K{0,1}, 1: K{2,3}
    const int l16  = lane & 15;

    const float* Xr = X + ((size_t)(b * Nn + bm * 16 + l16)) * DF + 2 * half;
    const float* Yr = Y + ((size_t)(b * Mm + bn * 16 + l16)) * DF + 2 * half;

    v8f acc = {};
#pragma unroll
    for (int kk = 0; kk < DF / 4; ++kk) {
        v2f a  = *(const v2f*)(Xr + kk * 4);
        v2f bv = *(const v2f*)(Yr + kk * 4);
        acc = __builtin_amdgcn_wmma_f32_16x16x4_f32(
            false, a, false, bv, (short)0, acc, false, false);
    }

    float* Drow = Dws + (size_t)b * Nn * Mm
                + (size_t)(bm * 16 + 8 * half) * Mm + bn * 16 + l16;
#pragma unroll
    for (int r = 0; r < 8; ++r) {
        Drow[(size_t)r * Mm] = -2.0f * acc[r];
    }
}

// ---------------------------------------------------------------------------
// Kernel 3: anti-diagonal DP.  One workgroup (512 threads = 16 wave32) per
// batch.  D is prefetched 8 diagonals at a time into a double-buffered,
// pad-9 skewed LDS tile via GLOBAL_LOAD_ASYNC_TO_LDS_B32 (ASYNCcnt), hiding
// stage-load latency behind the previous stage's compute.
// Each thread owns row n = t (cell i = t+1) and loads/reads only its own
// stage row, so a per-wave s_wait_asynccnt(0) is sufficient ordering.
// ---------------------------------------------------------------------------
__global__ __launch_bounds__(512) void softdtw_dp(
    const float* __restrict__ Dws, const float* __restrict__ xn,
    const float* __restrict__ yn, const int* __restrict__ Xl,
    const int* __restrict__ Yl, float* __restrict__ out)
{
    const int b = blockIdx.x;
    const int t = threadIdx.x;           // n = t, i = t+1

    __shared__ float stage[2][512 * SPAD];  // double-buffered D diagonals
    __shared__ float rbuf[3][Nn + 1];       // rotating R diagonals
    __shared__ float ynl[Mm];               // y row norms for this batch

    const float* Db = Dws + (size_t)b * Nn * Mm + (size_t)t * Mm;
    const float xnorm = xn[b * Nn + t];
    ynl[t] = yn[b * Mm + t];

    rbuf[0][t + 1] = BIGF;
    rbuf[1][t + 1] = BIGF;
    rbuf[2][t + 1] = BIGF;
    if (t == 0) { rbuf[0][0] = 0.0f; rbuf[1][0] = BIGF; rbuf[2][0] = BIGF; }

    const int tIdx = Xl[b] + Yl[b] - 2;  // target diagonal index (d-2)
    const int iOut = Xl[b] - 1;          // thread that owns the output cell

    const int NSTG = (Nn + Mm - 1 + SDEPTH - 1) / SDEPTH;  // 128 stages

    // Prologue: async-prefetch stage 0 (slot (t,k) <- D[t, k - t]).
    {
        const int mbase = -t;
#pragma unroll
        for (int k = 0; k < SDEPTH; ++k) {
            int m = mbase + k;
            if (m >= 0 && m < Mm)
                async_load_f32(lds_off(&stage[0][t * SPAD + k]), Db + m);
        }
    }

    int p2 = 0, p1 = 1, pc = 2;
    __syncthreads();

    for (int s = 0; s < NSTG; ++s) {
        // Wait for this wave's stage-s async loads (own rows only).
        wait_asynccnt0();

        // Prefetch stage s+1 into the other buffer; its previous contents
        // (stage s-1) were fully consumed before these issues (program order
        // per thread + per-diagonal barriers force the DS reads to complete).
        if (s + 1 < NSTG) {
            const int mbase = (s + 1) * SDEPTH - t;
#pragma unroll
            for (int k = 0; k < SDEPTH; ++k) {
                int m = mbase + k;
                if (m >= 0 && m < Mm)
                    async_load_f32(lds_off(&stage[(s + 1) & 1][t * SPAD + k]),
                                   Db + m);
            }
        }

        const float* st = &stage[s & 1][t * SPAD];

#pragma unroll 1
        for (int dd = 0; dd < SDEPTH; ++dd) {
            const int dIdx = s * SDEPTH + dd;
            if (dIdx > Nn + Mm - 2) break;

            const int m = dIdx - t;                 // j-1
            const bool valid = (m >= 0) && (m < Mm);

            const float z0 = rbuf[p2][t];           // R[i-1, j-1]
            const float z1 = rbuf[p1][t];           // R[i-1, j]
            const float z2 = rbuf[p1][t + 1];       // R[i,   j-1]
            const float mn = fminf(z0, fminf(z1, z2));
            const float sm = __expf(mn - z0) + __expf(mn - z1) + __expf(mn - z2);
            const float soft = mn - __logf(sm);     // gamma = 1

            const float Dval = st[dd] + xnorm + (valid ? ynl[m] : 0.0f);
            const float cur = valid ? (Dval + soft) : BIGF;

            rbuf[pc][t + 1] = cur;
            if (t == 0) rbuf[pc][0] = BIGF;
            if (dIdx == tIdx && t == iOut) out[b] = cur;
            __syncthreads();

            int tmp = p2; p2 = p1; p1 = pc; pc = tmp;
        }
    }
}

// ---------------------------------------------------------------------------
extern "C" void kernel_launch(void* const* d_in, const int* in_sizes, int n_in,
                              void* d_out, int out_size, void* d_ws, size_t ws_size,
                              hipStream_t stream) {
    (void)in_sizes; (void)n_in; (void)out_size; (void)ws_size;
    const float* X  = (const float*)d_in[0];
    const float* Y  = (const float*)d_in[1];
    const int*   Xl = (const int*)d_in[2];
    const int*   Yl = (const int*)d_in[3];
    float* out = (float*)d_out;

    // ws layout: D (B*N*M f32) | xnorm (B*N) | ynorm (B*M)  (~33.7 MB)
    float* Dws = (float*)d_ws;
    float* xn  = Dws + (size_t)Bb * Nn * Mm;
    float* yn  = xn + (size_t)Bb * Nn;

    softdtw_norms<<<(Bb * (Nn + Mm)) / 256, 256, 0, stream>>>(X, Y, xn, yn);

    dim3 ggrid(Mm / 64, Nn / 16, Bb);  // 4 waves/block, 1 tile/wave
    softdtw_gemm<<<ggrid, 128, 0, stream>>>(X, Y, Dws);

    softdtw_dp<<<Bb, 512, 0, stream>>>(Dws, xn, yn, Xl, Yl, out);
}